// Conv1DMHSA_38027640439361
// MI455X (gfx1250) — compile-verified
//
#include <hip/hip_runtime.h>
#include <math.h>

typedef float v2f __attribute__((ext_vector_type(2)));
typedef float v8f __attribute__((ext_vector_type(8)));

static constexpr int BATCH = 2;
static constexpr int H = 8;
static constexpr int D = 128;   // per-head dim (== channels)
static constexpr int R = 128;   // sketch dim
static constexpr int L = 2048;
static constexpr int O = 1024;  // H*D

#define WAVE 32
#define WPB  8
#define BLOCK (WAVE * WPB)

// D = A(16x4) * B(4x16) + C  in full fp32 (V_WMMA_F32_16X16X4_F32)
__device__ __forceinline__ v8f wmma4(v2f a, v2f b, v8f c) {
  return __builtin_amdgcn_wmma_f32_16x16x4_f32(false, a, false, b, (short)0, c,
                                               false, false);
}

// ===========================================================================
// Prep kernels: K-paired layouts so every WMMA fragment is a single b64 load.
// Paired layout for matrix M[K][N]: Mp[(k>>1)*2N + n*2 + (k&1)] = M[k][n].
// ===========================================================================

// xp (paired across channel c): xp[b][(c>>1)*2L + p*2 + (c&1)] = x[b][c][p]
__global__ void pair_x_kernel(const float* __restrict__ x, float* __restrict__ xp) {
  const int id = blockIdx.x * blockDim.x + threadIdx.x;  // [0, B*D*L)
  const int b   = id >> 18;                              // D*L = 2^18
  const int rem = id & (D * L - 1);
  const int e   = rem & 1;
  const int p   = (rem >> 1) & (L - 1);
  const int cp  = rem >> 12;
  xp[id] = x[(size_t)b * D * L + (size_t)(cp * 2 + e) * L + p];
}

// wTp[(k>>1)*2O + n*2 + (k&1)] = w[n, c, t],  k = t*D + c
__global__ void transpose_w_kernel(const float* __restrict__ w,  // [O, D, KS]
                                   float* __restrict__ wT, int KS) {
  const int id = blockIdx.x * blockDim.x + threadIdx.x;  // [0, KS*D*O)
  const int e  = id & 1;
  const int n  = (id >> 1) & (O - 1);
  const int kp = id >> 11;
  const int k  = kp * 2 + e;
  const int c  = k & (D - 1);
  const int t  = k >> 7;
  wT[id] = w[(size_t)n * (D * KS) + c * KS + t];
}

// gp[(k>>1)*2R + n*2 + (k&1)] = g[k*R + n]
__global__ void pair_g_kernel(const float* __restrict__ g, float* __restrict__ gp) {
  const int id = blockIdx.x * blockDim.x + threadIdx.x;  // [0, D*R)
  const int e  = id & 1;
  const int n  = (id >> 1) & (R - 1);
  const int kp = id >> 8;
  gp[id] = g[(size_t)(kp * 2 + e) * R + n];
}

// ===========================================================================
// Conv1d (causal pad) as GEMM, k = t*D + c, 16x32 tile (2 accumulators).
// ===========================================================================
template <int KS, bool GUARD>
__device__ __forceinline__ void conv_mac2(const float* __restrict__ xpb,  // paired x (this batch)
                                          const float* __restrict__ wTp,  // paired wT
                                          int l, int n0, int khalf,
                                          v8f& acc0, v8f& acc1) {
#pragma unroll
  for (int t = 0; t < KS; ++t) {
    const int p = l + t - (KS - 1);              // lane-dependent, c-invariant
    const int pc = GUARD ? (p < 0 ? 0 : p) : p;  // clamped (always-valid) address
    const float zmask = GUARD ? (p < 0 ? 0.0f : 1.0f) : 1.0f;
    const float* xt = xpb + 2 * pc;
    const float* wk = wTp + (size_t)(t * (D / 2)) * (2 * O) + 2 * n0;
#pragma unroll 8
    for (int c = 0; c < D; c += 4) {
      const int kp = (c + khalf) >> 1;
      v2f a = *(const v2f*)(xt + (size_t)kp * (2 * L));
      if (GUARD) { a.x *= zmask; a.y *= zmask; }
      const v2f b0 = *(const v2f*)(wk + (size_t)kp * (2 * O));
      const v2f b1 = *(const v2f*)(wk + (size_t)kp * (2 * O) + 32);
      acc0 = wmma4(a, b0, acc0);
      acc1 = wmma4(a, b1, acc1);
    }
  }
}

template <int KS, bool PAIRED_OUT>
__global__ void __launch_bounds__(BLOCK)
conv_gemm_kernel(const float* __restrict__ xp,     // [B][D/2, L, 2] paired
                 const float* __restrict__ wTp,    // [(KS*D)/2, O, 2] paired
                 const float* __restrict__ bias,   // [O] or null
                 const float* __restrict__ gamma,  // [1] or null
                 const float* __restrict__ beta,   // [1] or null
                 float* __restrict__ out) {        // [B,H,L,D] (row or l-paired)
  const int lane = threadIdx.x & 31;
  const int wave = threadIdx.x >> 5;
  const int tile = blockIdx.x * WPB + wave;
  const int O_T2 = O / 32, L_T = L / 16;
  const int o_t = tile % O_T2;
  const int l_t = (tile / O_T2) % L_T;
  const int b   = tile / (O_T2 * L_T);

  const int row   = lane & 15;
  const int khalf = (lane >> 4) << 1;  // 0 (lanes 0-15) or 2 (lanes 16-31)
  const int l     = l_t * 16 + row;
  const int n0    = o_t * 32 + row;

  const float* xpb = xp + (size_t)b * D * L;
  v8f acc0 = {}, acc1 = {};
  if (KS > 1 && l_t == 0) {  // wave-uniform: only first l-tile touches the pad
    conv_mac2<KS, true>(xpb, wTp, l, n0, khalf, acc0, acc1);
  } else {
    conv_mac2<KS, false>(xpb, wTp, l, n0, khalf, acc0, acc1);
  }

  const float g  = gamma ? gamma[0] : 1.0f;
  const float bt = beta ? beta[0] : 0.0f;
  const int col   = lane & 15;
  const int rbase = (lane >> 4) * 8;
#pragma unroll
  for (int j = 0; j < 2; ++j) {
    const v8f& acc = j ? acc1 : acc0;
    const int o_g = o_t * 32 + j * 16 + col;
    const int hh  = o_g >> 7;
    const int dd  = o_g & 127;
    const float bi = bias ? bias[o_g] : 0.0f;
    const size_t base = ((size_t)b * H + hh) * (size_t)L * D;
#pragma unroll
    for (int i = 0; i < 8; ++i) {
      const int ll = l_t * 16 + rbase + i;
      const size_t idx = PAIRED_OUT
          ? base + (size_t)(ll >> 1) * (2 * D) + dd * 2 + (ll & 1)  // l-paired
          : base + (size_t)ll * D + dd;                             // row-major
      out[idx] = g * (acc[i] + bi) + bt;
    }
  }
}

// ===========================================================================
// phi = sqrt(R) * tanh((q@g1)*(q@g2)/sqrt(R)); two WMMA GEMMs sharing A.
// ===========================================================================
template <bool PAIRED_OUT>
__global__ void __launch_bounds__(BLOCK)
sketch_kernel(const float* __restrict__ q,     // [BH, L, D] row-major
              const float* __restrict__ g1p,   // [D/2, R, 2] paired
              const float* __restrict__ g2p,   // [D/2, R, 2] paired
              float* __restrict__ phi) {       // [BH, L, R] (row or l-paired)
  const int lane = threadIdx.x & 31;
  const int wave = threadIdx.x >> 5;
  const int tile = blockIdx.x * WPB + wave;
  const int R_T = R / 16, L_T = L / 16;
  const int r_t = tile % R_T;
  const int l_t = (tile / R_T) % L_T;
  const int bh  = tile / (R_T * L_T);

  const int row   = lane & 15;
  const int khalf = (lane >> 4) << 1;
  const float* qrow = q + ((size_t)bh * L + l_t * 16 + row) * D + khalf;
  const float* g1b = g1p + 2 * (r_t * 16 + row);
  const float* g2b = g2p + 2 * (r_t * 16 + row);

  v8f c1 = {}, c2 = {};
#pragma unroll 8
  for (int k = 0; k < D; k += 4) {
    const int kp = (k + khalf) >> 1;
    const v2f a  = *(const v2f*)(qrow + k);
    const v2f b1 = *(const v2f*)(g1b + (size_t)kp * (2 * R));
    const v2f b2 = *(const v2f*)(g2b + (size_t)kp * (2 * R));
    c1 = wmma4(a, b1, c1);
    c2 = wmma4(a, b2, c2);
  }

  const float SQRT_R = 11.313708498984761f;      // sqrt(128)
  const float INV_SQRT_R = 0.08838834764831845f; // 1/sqrt(128)
  const int col   = lane & 15;
  const int rbase = (lane >> 4) * 8;
  const int rcol  = r_t * 16 + col;
  const size_t base = (size_t)bh * L * R;
#pragma unroll
  for (int i = 0; i < 8; ++i) {
    const int ll = l_t * 16 + rbase + i;
    const float s = c1[i] * c2[i] * INV_SQRT_R;
    const size_t idx = PAIRED_OUT
        ? base + (size_t)(ll >> 1) * (2 * R) + rcol * 2 + (ll & 1)
        : base + (size_t)ll * R + rcol;
    phi[idx] = SQRT_R * tanhf(s);
  }
}

// ===========================================================================
// KV[r,d] = sum_l phiK[l,r] * v[l,d]; 16x32 tile; K = L = 2048.
// phiK and v are l-paired; KV written r-paired.
// ===========================================================================
__global__ void __launch_bounds__(BLOCK)
kv_kernel(const float* __restrict__ phiKp,  // [BH][L/2, R, 2]
          const float* __restrict__ vp,     // [BH][L/2, D, 2]
          float* __restrict__ kvp) {        // [BH][R/2, D, 2]
  const int lane = threadIdx.x & 31;
  const int wave = threadIdx.x >> 5;
  const int tile = blockIdx.x * WPB + wave;
  const int D_T2 = D / 32, R_T = R / 16;
  const int d_t = tile % D_T2;
  const int r_t = (tile / D_T2) % R_T;
  const int bh  = tile / (D_T2 * R_T);

  const int row   = lane & 15;
  const int khalf = (lane >> 4) << 1;
  const float* pk = phiKp + (size_t)bh * L * R + 2 * (r_t * 16 + row);
  const float* vb = vp    + (size_t)bh * L * D + 2 * (d_t * 32 + row);

  v8f acc0 = {}, acc1 = {};
#pragma unroll 8
  for (int k = 0; k < L; k += 4) {
    const int kp = (k + khalf) >> 1;
    const v2f a  = *(const v2f*)(pk + (size_t)kp * (2 * R));
    const v2f b0 = *(const v2f*)(vb + (size_t)kp * (2 * D));
    const v2f b1 = *(const v2f*)(vb + (size_t)kp * (2 * D) + 32);
    acc0 = wmma4(a, b0, acc0);
    acc1 = wmma4(a, b1, acc1);
  }
  const int col   = lane & 15;
  const int rbase = (lane >> 4) * 8;
  const size_t base = (size_t)bh * R * D;
#pragma unroll
  for (int j = 0; j < 2; ++j) {
    const v8f& acc = j ? acc1 : acc0;
    const int dd = d_t * 32 + j * 16 + col;
#pragma unroll
    for (int i = 0; i < 8; ++i) {
      const int rg = r_t * 16 + rbase + i;
      kvp[base + (size_t)(rg >> 1) * (2 * D) + dd * 2 + (rg & 1)] = acc[i];
    }
  }
}

// ksum[bh,r] = sum_l phiK[bh,l,r]  (phiK is l-paired)
__global__ void ksum_kernel(const float* __restrict__ phiKp, float* __restrict__ ksum) {
  const int id = blockIdx.x * blockDim.x + threadIdx.x;  // [0, BH*R)
  const int bh = id / R, r = id % R;
  const float* p = phiKp + (size_t)bh * L * R + 2 * r;
  float s = 0.0f;
#pragma unroll 8
  for (int lp = 0; lp < L / 2; ++lp) {
    const v2f t = *(const v2f*)(p + (size_t)lp * (2 * R));
    s += t.x + t.y;
  }
  ksum[id] = s;
}

// den[bh,l] = phiQ[bh,l,:] . ksum[bh,:] + 1e-6   (phiQ row-major)
__global__ void den_kernel(const float* __restrict__ phiQ,
                           const float* __restrict__ ksum,
                           float* __restrict__ den) {
  const int id = blockIdx.x * blockDim.x + threadIdx.x;  // [0, BH*L)
  const int bh = id / L;
  const float4* p  = (const float4*)(phiQ + (size_t)id * R);
  const float4* ks = (const float4*)(ksum + (size_t)bh * R);
  float s = 0.0f;
#pragma unroll 8
  for (int r = 0; r < R / 4; ++r) {
    const float4 a = p[r], b = ks[r];
    s += a.x * b.x + a.y * b.y + a.z * b.z + a.w * b.w;
  }
  den[id] = s + 1e-6f;
}

// ===========================================================================
// o[l,d] = (phiQ[l,:] @ KV[:,d]) / den[l]; 16x32 tile; KV is r-paired.
// Output o_flat[b, l, h*D+d] row-major (A-side of the projection GEMM).
// ===========================================================================
__global__ void __launch_bounds__(BLOCK)
attn_out_kernel(const float* __restrict__ phiQ,  // [BH, L, R] row-major
                const float* __restrict__ kvp,   // [BH][R/2, D, 2]
                const float* __restrict__ den,   // [BH, L]
                float* __restrict__ o_flat) {    // [B, L, O]
  const int lane = threadIdx.x & 31;
  const int wave = threadIdx.x >> 5;
  const int tile = blockIdx.x * WPB + wave;
  const int D_T2 = D / 32, L_T = L / 16;
  const int d_t = tile % D_T2;
  const int l_t = (tile / D_T2) % L_T;
  const int bh  = tile / (D_T2 * L_T);
  const int b = bh >> 3, h = bh & 7;

  const int row   = lane & 15;
  const int khalf = (lane >> 4) << 1;
  const float* qrow = phiQ + ((size_t)bh * L + l_t * 16 + row) * R + khalf;
  const float* kvb  = kvp + (size_t)bh * R * D + 2 * (d_t * 32 + row);

  v8f acc0 = {}, acc1 = {};
#pragma unroll 8
  for (int k = 0; k < R; k += 4) {
    const int kp = (k + khalf) >> 1;
    const v2f a  = *(const v2f*)(qrow + k);
    const v2f b0 = *(const v2f*)(kvb + (size_t)kp * (2 * D));
    const v2f b1 = *(const v2f*)(kvb + (size_t)kp * (2 * D) + 32);
    acc0 = wmma4(a, b0, acc0);
    acc1 = wmma4(a, b1, acc1);
  }
  const int col   = lane & 15;
  const int rbase = (lane >> 4) * 8;
#pragma unroll
  for (int i = 0; i < 8; ++i) {
    const int ll = l_t * 16 + rbase + i;
    const float inv = 1.0f / den[(size_t)bh * L + ll];
    const size_t base = ((size_t)b * L + ll) * O + h * D + d_t * 32;
    o_flat[base + col]      = acc0[i] * inv;
    o_flat[base + 16 + col] = acc1[i] * inv;
  }
}

// ===========================================================================
// out[m,j] = sum_i o_flat[m,i] * proj_w[j,i] + proj_b[j]; 16x32 tile.
// ===========================================================================
__global__ void __launch_bounds__(BLOCK)
proj_kernel(const float* __restrict__ o_flat,  // [B*L, O]
            const float* __restrict__ pw,      // [D, O]
            const float* __restrict__ pb,      // [D]
            float* __restrict__ out) {         // [B*L, D]
  const int lane = threadIdx.x & 31;
  const int wave = threadIdx.x >> 5;
  const int tile = blockIdx.x * WPB + wave;
  const int N_T2 = D / 32;
  const int n_t = tile % N_T2;
  const int m_t = tile / N_T2;

  const int row   = lane & 15;
  const int khalf = (lane >> 4) << 1;
  const float* arow  = o_flat + ((size_t)m_t * 16 + row) * O + khalf;
  const float* brow0 = pw + ((size_t)(n_t * 32 + row)) * O + khalf;
  const float* brow1 = pw + ((size_t)(n_t * 32 + 16 + row)) * O + khalf;

  v8f acc0 = {}, acc1 = {};
#pragma unroll 8
  for (int k = 0; k < O; k += 4) {
    const v2f a  = *(const v2f*)(arow + k);
    const v2f b0 = *(const v2f*)(brow0 + k);
    const v2f b1 = *(const v2f*)(brow1 + k);
    acc0 = wmma4(a, b0, acc0);
    acc1 = wmma4(a, b1, acc1);
  }
  const int col   = lane & 15;
  const int rbase = (lane >> 4) * 8;
  const float bias0 = pb[n_t * 32 + col];
  const float bias1 = pb[n_t * 32 + 16 + col];
#pragma unroll
  for (int i = 0; i < 8; ++i) {
    const int m = m_t * 16 + rbase + i;
    out[(size_t)m * D + n_t * 32 + col]      = acc0[i] + bias0;
    out[(size_t)m * D + n_t * 32 + 16 + col] = acc1[i] + bias1;
  }
}

// ---------------------------------------------------------------------------
extern "C" void kernel_launch(void* const* d_in, const int* in_sizes, int n_in,
                              void* d_out, int out_size, void* d_ws, size_t ws_size,
                              hipStream_t stream) {
  const float* x       = (const float*)d_in[0];
  const float* q_w     = (const float*)d_in[1];
  const float* q_b     = (const float*)d_in[2];
  const float* k_w     = (const float*)d_in[3];
  const float* k_b     = (const float*)d_in[4];
  const float* v_w     = (const float*)d_in[5];
  const float* g1_q    = (const float*)d_in[6];
  const float* g2_q    = (const float*)d_in[7];
  const float* g1_k    = (const float*)d_in[8];
  const float* g2_k    = (const float*)d_in[9];
  const float* gamma_q = (const float*)d_in[10];
  const float* beta_q  = (const float*)d_in[11];
  const float* gamma_k = (const float*)d_in[12];
  const float* beta_k  = (const float*)d_in[13];
  const float* proj_w  = (const float*)d_in[14];
  const float* proj_b  = (const float*)d_in[15];
  float* out = (float*)d_out;

  float* ws = (float*)d_ws;
  const size_t S = (size_t)BATCH * H * L * D;  // 4,194,304 floats
  float* qk_buf = ws;                               // q/k row-major; reused as o_flat
  float* phi_q  = ws + S;                           // row-major
  float* phi_k  = ws + 2 * S;                       // l-paired
  float* v_buf  = ws + 3 * S;                       // l-paired
  float* kvb    = ws + 4 * S;                       // [BH,R,D] r-paired 262,144
  float* ksb    = kvb + (size_t)BATCH * H * R * D;  // [BH,R]    2,048
  float* denb   = ksb + (size_t)BATCH * H * R;      // [BH,L]    32,768
  float* wTb    = denb + (size_t)BATCH * H * L;     // [3*D,O] paired 393,216
  float* xpb    = wTb + (size_t)3 * D * O;          // paired x 524,288
  float* g1qp   = xpb + (size_t)BATCH * D * L;      // 16,384 each
  float* g2qp   = g1qp + (size_t)D * R;
  float* g1kp   = g2qp + (size_t)D * R;
  float* g2kp   = g1kp + (size_t)D * R;

  const dim3 blk(BLOCK);
  const int conv_tiles = BATCH * (L / 16) * (O / 32);      // 8192
  const int sk_tiles   = BATCH * H * (L / 16) * (R / 16);  // 16384
  const int kv_tiles   = BATCH * H * (R / 16) * (D / 32);  // 512
  const int o_tiles    = BATCH * H * (L / 16) * (D / 32);  // 8192
  const int pj_tiles   = (BATCH * L / 16) * (D / 32);      // 1024

  // Layout prep
  pair_x_kernel<<<(BATCH * D * L) / 256, 256, 0, stream>>>(x, xpb);
  pair_g_kernel<<<(D * R) / 256, 256, 0, stream>>>(g1_q, g1qp);
  pair_g_kernel<<<(D * R) / 256, 256, 0, stream>>>(g2_q, g2qp);
  pair_g_kernel<<<(D * R) / 256, 256, 0, stream>>>(g1_k, g1kp);
  pair_g_kernel<<<(D * R) / 256, 256, 0, stream>>>(g2_k, g2kp);

  // Q path
  transpose_w_kernel<<<(3 * D * O) / 256, 256, 0, stream>>>(q_w, wTb, 3);
  conv_gemm_kernel<3, false><<<conv_tiles / WPB, blk, 0, stream>>>(xpb, wTb, q_b, gamma_q, beta_q, qk_buf);
  sketch_kernel<false><<<sk_tiles / WPB, blk, 0, stream>>>(qk_buf, g1qp, g2qp, phi_q);
  // K path (reuse qk_buf, wTb)
  transpose_w_kernel<<<(3 * D * O) / 256, 256, 0, stream>>>(k_w, wTb, 3);
  conv_gemm_kernel<3, false><<<conv_tiles / WPB, blk, 0, stream>>>(xpb, wTb, k_b, gamma_k, beta_k, qk_buf);
  sketch_kernel<true><<<sk_tiles / WPB, blk, 0, stream>>>(qk_buf, g1kp, g2kp, phi_k);
  // V path (kernel size 1, no bias/affine), l-paired output
  transpose_w_kernel<<<(D * O) / 256, 256, 0, stream>>>(v_w, wTb, 1);
  conv_gemm_kernel<1, true><<<conv_tiles / WPB, blk, 0, stream>>>(xpb, wTb, nullptr, nullptr, nullptr, v_buf);
  // Linear attention
  kv_kernel<<<kv_tiles / WPB, blk, 0, stream>>>(phi_k, v_buf, kvb);
  ksum_kernel<<<(BATCH * H * R) / 256, 256, 0, stream>>>(phi_k, ksb);
  den_kernel<<<(BATCH * H * L) / 256, 256, 0, stream>>>(phi_q, ksb, denb);
  attn_out_kernel<<<o_tiles / WPB, blk, 0, stream>>>(phi_q, kvb, denb, qk_buf);
  // Final projection
  proj_kernel<<<pj_tiles / WPB, blk, 0, stream>>>(qk_buf, proj_w, proj_b, out);
}